// GIN_Model_31662498906707
// MI455X (gfx1250) — compile-verified
//
#include <hip/hip_runtime.h>
#include <hip/hip_bf16.h>

// MI455X / gfx1250, wave32. fp32 WMMA (16x16x4) fused GAT layer, v2:
//  - 32-query tiles (8 waves / 256 threads) -> halves Whk L2 re-read traffic
//  - a_w * N folded into the query LDS tile (removes VALU from K-loop)
//  - op_W stored transposed in LDS (conflict-free gate dot)

#define BB   32
#define NN   512
#define FF   128
#define DOP  48

typedef __attribute__((ext_vector_type(2))) float v2f;
typedef __attribute__((ext_vector_type(8))) float v8f;

__device__ __forceinline__ float wave_sum(float v) {
    v += __shfl_xor(v, 16, 32);
    v += __shfl_xor(v, 8, 32);
    v += __shfl_xor(v, 4, 32);
    v += __shfl_xor(v, 2, 32);
    v += __shfl_xor(v, 1, 32);
    return v;
}
__device__ __forceinline__ float wave_max(float v) {
    v = fmaxf(v, __shfl_xor(v, 16, 32));
    v = fmaxf(v, __shfl_xor(v, 8, 32));
    v = fmaxf(v, __shfl_xor(v, 4, 32));
    v = fmaxf(v, __shfl_xor(v, 2, 32));
    v = fmaxf(v, __shfl_xor(v, 1, 32));
    return v;
}

// ---------------------------------------------------------------------------
// Kernel 1: Whk[b,n,o] = sum_f h[b,n,f] * Wk[o,f]
// One 16x16 output tile per wave; fp32 WMMA layout:
//   lane = 16*hh + n ; A VGPR j -> A[M=n, K=j+2*hh]
//                      B VGPR j -> B[K=j+2*hh, N=n]
//                      D VGPR j -> D[M=j+8*hh, N=n]
// ---------------------------------------------------------------------------
__global__ void __launch_bounds__(128) whk_kernel(const float* __restrict__ h,
                                                  const float* __restrict__ Wk,
                                                  float* __restrict__ whk) {
    const int wave = threadIdx.x >> 5;
    const int lane = threadIdx.x & 31;
    const int n  = lane & 15;
    const int hh = lane >> 4;
    const int tile = blockIdx.x * 4 + wave;   // 8192 tiles
    const int r0 = (tile >> 3) * 16;          // flattened row tile (b*N+n)
    const int o0 = (tile & 7) * 16;           // output-feature tile

    v8f c = {};
    const float* arow = h  + (size_t)(r0 + n) * FF + 2 * hh;
    const float* brow = Wk + (size_t)(o0 + n) * FF + 2 * hh;
    for (int m0 = 0; m0 < FF; m0 += 4) {
        v2f a, b;
        a.x = arow[m0];     a.y = arow[m0 + 1];
        b.x = brow[m0];     b.y = brow[m0 + 1];
        c = __builtin_amdgcn_wmma_f32_16x16x4_f32(false, a, false, b,
                                                  (short)0, c, false, false);
    }
#pragma unroll
    for (int j = 0; j < 8; ++j)
        whk[(size_t)(r0 + j + 8 * hh) * FF + o0 + n] = c[j];
}

// ---------------------------------------------------------------------------
// Kernel 2: fused scores -> leakyrelu*adj -> softmax -> attn@Whk -> gate -> LN
// One block (256 thr / 8 waves) per (batch b, 32-query tile).
// ---------------------------------------------------------------------------
__global__ void __launch_bounds__(256) attn_kernel(
    const float* __restrict__ whk,   const float* __restrict__ adj,
    const float* __restrict__ op_emb,const float* __restrict__ a_w,
    const float* __restrict__ op_W,  const float* __restrict__ op_b,
    const float* __restrict__ gamma, const float* __restrict__ beta,
    float* __restrict__ out) {

    __shared__ float sWq[32][FF];        // query Whk * (a_w*N)      16 KB
    __shared__ float sS [32][NN];        // score / attn rows        64 KB
    __shared__ float sH [32][FF];        // h' accumulator           16 KB
    __shared__ float sOpE[32][DOP];      // op_emb rows               6 KB
    __shared__ float sOpWT[DOP][FF];     // op_W transposed          24 KB
    __shared__ float sOpB[FF], sG[FF], sBt[FF];

    const int tid  = threadIdx.x;
    const int wave = tid >> 5;           // 0..7
    const int lane = tid & 31;
    const int n  = lane & 15;
    const int hh = lane >> 4;
    const int b  = blockIdx.x >> 4;      // 0..31
    const int q0 = (blockIdx.x & 15) * 32;
    const float* whk_b = whk + (size_t)b * NN * FF;

    // ---- Phase 1: LDS preload -------------------------------------------
    for (int i = tid; i < 32 * FF; i += 256) {
        const int r = i >> 7, cc = i & 127;
        // fold a_w * N into the query tile (LeakyReLU is positively homogeneous,
        // so the scale must survive into softmax; queries are only used scaled here)
        sWq[r][cc] = whk_b[(size_t)(q0 + r) * FF + cc] * (a_w[cc] * (float)NN);
    }
    for (int i = tid; i < FF; i += 256) {
        sOpB[i] = op_b[i]; sG[i] = gamma[i]; sBt[i] = beta[i];
    }
    for (int i = tid; i < 32 * DOP; i += 256)
        sOpE[i / DOP][i % DOP] = op_emb[((size_t)b * NN + q0 + i / DOP) * DOP + i % DOP];
    for (int i = tid; i < DOP * FF; i += 256) {
        const int d = i >> 7, cc = i & 127;
        sOpWT[d][cc] = op_W[cc * DOP + d];
    }
    __syncthreads();

    // ---- Phase 2: scores, leakyrelu, adj gate ---------------------------
    // 64 jobs = 2 query subtiles x 32 key tiles; 8 jobs per wave
    for (int job = wave; job < 64; job += 8) {
        const int qs = (job & 1) * 16;        // query subtile base row
        const int l0 = (job >> 1) * 16;       // key tile base
        v8f c = {};
        const float* krow = whk_b + (size_t)(l0 + n) * FF + 2 * hh;
        for (int m0 = 0; m0 < FF; m0 += 4) {
            v2f a, bb;
            a.x  = sWq[qs + n][m0 + 2 * hh];
            a.y  = sWq[qs + n][m0 + 2 * hh + 1];
            bb.x = krow[m0];
            bb.y = krow[m0 + 1];
            c = __builtin_amdgcn_wmma_f32_16x16x4_f32(false, a, false, bb,
                                                      (short)0, c, false, false);
        }
#pragma unroll
        for (int j = 0; j < 8; ++j) {
            const int e = qs + j + 8 * hh;                // query row in 32-tile
            float s = c[j];
            s = (s > 0.0f) ? s : 0.2f * s;                // LeakyReLU(0.2)
            s *= adj[((size_t)b * NN + q0 + e) * NN + l0 + n];
            sS[e][l0 + n] = s;
        }
    }
    __syncthreads();

    // ---- Phase 3: softmax over key axis (4 rows per wave) ---------------
    for (int r = wave * 4; r < wave * 4 + 4; ++r) {
        float vals[NN / 32];
        float mx = -1e30f;
#pragma unroll
        for (int i = 0; i < NN / 32; ++i) {
            vals[i] = sS[r][lane + 32 * i];
            mx = fmaxf(mx, vals[i]);
        }
        mx = wave_max(mx);
        float sum = 0.0f;
#pragma unroll
        for (int i = 0; i < NN / 32; ++i) {
            vals[i] = __expf(vals[i] - mx);
            sum += vals[i];
        }
        sum = wave_sum(sum);
        const float inv = 1.0f / sum;
#pragma unroll
        for (int i = 0; i < NN / 32; ++i)
            sS[r][lane + 32 * i] = vals[i] * inv;
    }
    __syncthreads();

    // ---- Phase 4: h' = attn[32,512] @ Whk[512,128] ----------------------
    // 16 output tiles = 2 query subtiles x 8 col tiles; 2 per wave
#pragma unroll
    for (int t = 0; t < 2; ++t) {
        const int tt = wave * 2 + t;          // 0..15
        const int qs = (tt >> 3) * 16;        // query subtile base row
        const int o0 = (tt & 7) * 16;         // output-feature tile
        v8f c = {};
        for (int l0 = 0; l0 < NN; l0 += 4) {
            v2f a, bb;
            a.x  = sS[qs + n][l0 + 2 * hh];
            a.y  = sS[qs + n][l0 + 2 * hh + 1];
            bb.x = whk_b[(size_t)(l0 + 2 * hh) * FF + o0 + n];
            bb.y = whk_b[(size_t)(l0 + 2 * hh + 1) * FF + o0 + n];
            c = __builtin_amdgcn_wmma_f32_16x16x4_f32(false, a, false, bb,
                                                      (short)0, c, false, false);
        }
#pragma unroll
        for (int j = 0; j < 8; ++j)
            sH[qs + j + 8 * hh][o0 + n] = c[j];
    }
    __syncthreads();

    // ---- Phase 5: sigmoid gate + LayerNorm (4 rows per wave) ------------
    for (int r = wave * 4; r < wave * 4 + 4; ++r) {
        float v[FF / 32];
        float ssum = 0.0f;
#pragma unroll
        for (int k = 0; k < FF / 32; ++k) {
            const int col = lane + 32 * k;
            float g = sOpB[col];
            for (int d = 0; d < DOP; ++d)
                g = fmaf(sOpE[r][d], sOpWT[d][col], g);   // conflict-free
            g = 1.0f / (1.0f + __expf(-g));
            v[k] = g * sH[r][col];
            ssum += v[k];
        }
        const float mean = wave_sum(ssum) * (1.0f / (float)FF);
        float vsum = 0.0f;
#pragma unroll
        for (int k = 0; k < FF / 32; ++k) {
            const float d = v[k] - mean;
            vsum += d * d;
        }
        const float var  = wave_sum(vsum) * (1.0f / (float)FF);
        const float rstd = rsqrtf(var + 1e-5f);
#pragma unroll
        for (int k = 0; k < FF / 32; ++k) {
            const int col = lane + 32 * k;
            out[((size_t)b * NN + q0 + r) * FF + col] =
                sG[col] * (v[k] - mean) * rstd + sBt[col];
        }
    }
}

extern "C" void kernel_launch(void* const* d_in, const int* in_sizes, int n_in,
                              void* d_out, int out_size, void* d_ws, size_t ws_size,
                              hipStream_t stream) {
    (void)in_sizes; (void)n_in; (void)out_size; (void)ws_size;
    const float* h      = (const float*)d_in[0];
    const float* adj    = (const float*)d_in[1];
    const float* op_emb = (const float*)d_in[2];
    const float* Wk     = (const float*)d_in[3];
    const float* a_w    = (const float*)d_in[4];
    const float* op_W   = (const float*)d_in[5];
    const float* op_b   = (const float*)d_in[6];
    const float* gamma  = (const float*)d_in[7];
    const float* beta   = (const float*)d_in[8];
    float* out = (float*)d_out;
    float* whk = (float*)d_ws;   // 32*512*128*4 = 8 MB scratch

    // Whk projection: 8192 16x16 tiles, 4 waves (tiles) per block
    whk_kernel<<<2048, 128, 0, stream>>>(h, Wk, whk);
    // Fused attention + gate + layernorm: one block per (b, 32-query tile)
    attn_kernel<<<BB * (NN / 32), 256, 0, stream>>>(whk, adj, op_emb, a_w,
                                                    op_W, op_b, gamma, beta, out);
}